// TargetAttention_30142080483365
// MI455X (gfx1250) — compile-verified
//
#include <hip/hip_runtime.h>

typedef __attribute__((ext_vector_type(2))) float v2f;
typedef __attribute__((ext_vector_type(4))) float v4f;
typedef __attribute__((ext_vector_type(8))) float v8f;

#define HEADS 8
#define EDIM  128

// Extract element `sel` (0..7) from a v8f with constant-index cndmask chain
// (avoids dynamic-index scratch traffic).
__device__ __forceinline__ float extract8(v8f a, int sel) {
    float r = a[0];
#pragma unroll
    for (int i = 1; i < 8; ++i) r = (sel == i) ? a[i] : r;
    return r;
}

__global__ __launch_bounds__(256) void target_attn_wmma(
    const float* __restrict__ q,    // [B, 1, 128]
    const float* __restrict__ kv,   // [B, 8, 128]
    float* __restrict__ out,        // [B, 128]
    int B)
{
    const int lane  = threadIdx.x & 31;
    const int wave  = blockIdx.x * (blockDim.x >> 5) + (threadIdx.x >> 5);
    const int nwave = gridDim.x * (blockDim.x >> 5);
    const int row   = lane & 15;                 // M (A) / N (B) index in tile
    const int koff  = (lane < 16) ? 0 : 2;       // K sub-offset held by this lane

    const int ntiles = B >> 4;                   // 16 batches per tile

    for (int tile = wave; tile < ntiles; tile += nwave) {
        const int b0 = tile << 4;

        // Per-lane base pointers (element offsets fit size_t; bytes ~0.5GB)
        const float* qp  = q  + (size_t)(b0 + row) * EDIM + koff;
        const float* kvp = kv + (size_t)(b0 + row) * HEADS * EDIM + koff;

        v8f acc[HEADS];
#pragma unroll
        for (int h = 0; h < HEADS; ++h) acc[h] = (v8f)0.0f;

        // Accumulate scores: C_h[i,j] += kv[b0+i,h,4c+k] * q[b0+j,4c+k]
        // 32 chunks of K=4 cover E=128. Each kv/q element is loaded once.
#pragma unroll 4
        for (int c = 0; c < EDIM / 4; ++c) {
            v2f bq = *(const v2f*)(qp + 4 * c);
#pragma unroll
            for (int h = 0; h < HEADS; ++h) {
                v2f a = *(const v2f*)(kvp + h * EDIM + 4 * c);
                acc[h] = __builtin_amdgcn_wmma_f32_16x16x4_f32(
                    /*neg_a=*/false, a, /*neg_b=*/false, bq,
                    /*c_mod=*/(short)0, acc[h],
                    /*reuse_a=*/false, /*reuse_b=*/false);
            }
        }

        // Diagonal extraction: batch i<8 lives in lane i (VGPR i),
        // batch i>=8 lives in lane i+16 (VGPR i-8).
        const bool owner = (lane < 8) || (lane >= 24);
        const int  sel   = (lane < 8) ? lane : ((lane >= 24) ? (lane - 24) : 0);

        int   hmax = 0;
        float smax = extract8(acc[0], sel);
#pragma unroll
        for (int h = 1; h < HEADS; ++h) {
            float s = extract8(acc[h], sel);
            if (s > smax) { smax = s; hmax = h; }   // strict '>' keeps first max
        }
        (void)owner;

        // Cooperative winner-row gather: 32 lanes copy 128 floats (b128 each).
#pragma unroll
        for (int i = 0; i < 16; ++i) {
            int src_lane = (i < 8) ? i : (i + 16);
            int hsel = __shfl(hmax, src_lane, 32);
            const v4f* sp = (const v4f*)(kv + ((size_t)(b0 + i) * HEADS + hsel) * EDIM) + lane;
            v4f val = *sp;
            *((v4f*)(out + (size_t)(b0 + i) * EDIM) + lane) = val;
        }
    }
}

extern "C" void kernel_launch(void* const* d_in, const int* in_sizes, int n_in,
                              void* d_out, int out_size, void* d_ws, size_t ws_size,
                              hipStream_t stream) {
    const float* q  = (const float*)d_in[0];   // [B,1,128] f32
    const float* kv = (const float*)d_in[1];   // [B,8,128] f32
    float* out = (float*)d_out;                // [B,128] f32

    int B = in_sizes[0] / EDIM;                // 131072
    int ntiles = B >> 4;                       // 8192 tiles (16 batches each)
    int waves_per_block = 256 / 32;            // 8
    int blocks = (ntiles + waves_per_block - 1) / waves_per_block;  // 1024

    target_attn_wmma<<<blocks, 256, 0, stream>>>(q, kv, out, B);
}